// DDGCRN_53609781789010
// MI455X (gfx1250) — compile-verified
//
#include <hip/hip_runtime.h>
#include <math.h>

constexpr int B_  = 8;
constexpr int T_  = 12;
constexpr int N_  = 716;
constexpr int E_  = 10;
constexpr int H_  = 64;
constexpr int DIN = 65;    // 1 + H
constexpr int NP  = 720;   // N padded to multiple of 16
constexpr int KP  = 80;    // DIN padded to multiple of 16
constexpr int EP  = 12;    // E padded to multiple of 4
constexpr int OG  = 128;   // gate GCN output (2H)
constexpr int OU  = 64;    // update GCN output (H)
constexpr int CHK = 48;    // K-chunk staged in LDS for k_xg2 (720 = 15*48)

typedef float v2f __attribute__((ext_vector_type(2)));
typedef float v8f __attribute__((ext_vector_type(8)));

__device__ __forceinline__ float sigmoidf_(float x) { return 1.0f / (1.0f + __expf(-x)); }

// ---------------------------------------------------------------- utilities
__global__ void k_zero(float* p, long n) {
    long i = (long)blockIdx.x * blockDim.x + threadIdx.x;
    if (i < n) p[i] = 0.0f;
}

// emb[b,t,n,e] = node_emb[n,e] * tid_emb[idx,e] * diw_emb[idx,e]
__global__ void k_embed(const float* __restrict__ src, const float* __restrict__ nodeE,
                        const float* __restrict__ tidE, const float* __restrict__ diwE,
                        float* __restrict__ emb) {
    long idx = (long)blockIdx.x * blockDim.x + threadIdx.x;  // over B*T*N
    if (idx >= (long)B_ * T_ * N_) return;
    int n = (int)(idx % N_);
    const float* s = src + idx * 3;
    int ti = (int)(s[1] * 288.0f); ti = ti < 0 ? 0 : (ti > 287 ? 287 : ti);
    int di = (int)(s[2]);          di = di < 0 ? 0 : (di > 6 ? 6 : di);
    float* o = emb + idx * E_;
    for (int e = 0; e < E_; ++e)
        o[e] = nodeE[n * E_ + e] * tidE[ti * E_ + e] * diwE[di * E_ + e];
}

// cur[t,b,n] = source[b,t,n,0]
__global__ void k_init_cur(const float* __restrict__ src, float* __restrict__ cur) {
    long idx = (long)blockIdx.x * blockDim.x + threadIdx.x;  // over T*B*N
    if (idx >= (long)T_ * B_ * N_) return;
    int n = (int)(idx % N_);
    int b = (int)((idx / N_) % B_);
    int t = (int)(idx / ((long)N_ * B_));
    cur[idx] = src[(((long)b * T_ + t) * N_ + n) * 3 + 0];
}

// W[n,k,i(KP),o] = sum_e static[n,e] * wp[e,k,i,o]   (i >= DIN -> 0)
__global__ void k_node_w(const float* __restrict__ st, const float* __restrict__ wp,
                         float* __restrict__ W, int O) {
    long idx = (long)blockIdx.x * blockDim.x + threadIdx.x;
    long tot = (long)N_ * 2 * KP * O;
    if (idx >= tot) return;
    int o = (int)(idx % O); long r = idx / O;
    int i = (int)(r % KP);  r /= KP;
    int k = (int)(r % 2);
    int n = (int)(r / 2);
    float acc = 0.0f;
    if (i < DIN)
        for (int e = 0; e < E_; ++e)
            acc += st[n * E_ + e] * wp[(((long)e * 2 + k) * DIN + i) * O + o];
    W[idx] = acc;
}

// Bv[n,o] = sum_e static[n,e] * bp[e,o]
__global__ void k_node_b(const float* __restrict__ st, const float* __restrict__ bp,
                         float* __restrict__ Bv, int O) {
    long idx = (long)blockIdx.x * blockDim.x + threadIdx.x;
    if (idx >= (long)N_ * O) return;
    int o = (int)(idx % O);
    int n = (int)(idx / O);
    float acc = 0.0f;
    for (int e = 0; e < E_; ++e) acc += st[n * E_ + e] * bp[e * O + o];
    Bv[idx] = acc;
}

// xs[b,n(NP),i(KP)]: i==0 -> x ; i in [1,64] -> state (mode0) or z*state (mode1) ; else 0
__global__ void k_build_xs(const float* __restrict__ curx, const float* __restrict__ state,
                           const float* __restrict__ zr, int mode, float* __restrict__ xs) {
    long idx = (long)blockIdx.x * blockDim.x + threadIdx.x;  // B*NP*KP
    if (idx >= (long)B_ * NP * KP) return;
    int i = (int)(idx % KP);
    int n = (int)((idx / KP) % NP);
    int b = (int)(idx / ((long)KP * NP));
    float v = 0.0f;
    if (n < N_ && i < DIN) {
        if (i == 0) v = curx[b * N_ + n];
        else {
            float s = state[((long)b * N_ + n) * H_ + (i - 1)];
            if (mode) v = zr[((long)b * N_ + n) * OG + (i - 1)] * s;
            else v = s;
        }
    }
    xs[idx] = v;
}

// FC chain + nodevec = tanh(emb * x_)  ->  nv[b,n(NP),e(EP)]  (pads zeroed)
__global__ void k_fc(const float* __restrict__ xs, const float* __restrict__ emb, int t,
                     const float* __restrict__ fc1W, const float* __restrict__ fc1b,
                     const float* __restrict__ fc2W, const float* __restrict__ fc2b,
                     const float* __restrict__ fc3W, const float* __restrict__ fc3b,
                     float* __restrict__ nv) {
    __shared__ float sW1[16 * DIN];
    __shared__ float sb1[16];
    __shared__ float sW2[2 * 16];
    __shared__ float sb2[2];
    __shared__ float sW3[E_ * 2];
    __shared__ float sb3[E_];
    for (int i = threadIdx.x; i < 16 * DIN; i += blockDim.x) sW1[i] = fc1W[i];
    for (int i = threadIdx.x; i < 16;       i += blockDim.x) sb1[i] = fc1b[i];
    for (int i = threadIdx.x; i < 32;       i += blockDim.x) sW2[i] = fc2W[i];
    for (int i = threadIdx.x; i < 2;        i += blockDim.x) sb2[i] = fc2b[i];
    for (int i = threadIdx.x; i < E_ * 2;   i += blockDim.x) sW3[i] = fc3W[i];
    for (int i = threadIdx.x; i < E_;       i += blockDim.x) sb3[i] = fc3b[i];
    __syncthreads();

    int idx = blockIdx.x * blockDim.x + threadIdx.x;  // B*NP
    if (idx >= B_ * NP) return;
    int n = idx % NP, b = idx / NP;
    float* onv = nv + ((long)b * NP + n) * EP;
    if (n >= N_) { for (int e = 0; e < EP; ++e) onv[e] = 0.0f; return; }

    const float* xr = xs + ((long)b * NP + n) * KP;
    float h1[16];
    for (int j = 0; j < 16; ++j) {
        float a = sb1[j];
        for (int i = 0; i < DIN; ++i) a += xr[i] * sW1[j * DIN + i];
        h1[j] = sigmoidf_(a);
    }
    float h2[2];
    for (int j = 0; j < 2; ++j) {
        float a = sb2[j];
        for (int i = 0; i < 16; ++i) a += h1[i] * sW2[j * 16 + i];
        h2[j] = sigmoidf_(a);
    }
    const float* er = emb + (((long)b * T_ + t) * N_ + n) * E_;
    for (int e = 0; e < E_; ++e) {
        float a = sb3[e] + h2[0] * sW3[e * 2 + 0] + h2[1] * sW3[e * 2 + 1];
        onv[e] = tanhf(er[e] * a);
    }
    onv[10] = 0.0f; onv[11] = 0.0f;
}

// graph[b,n,m] = relu(nv[b,n,:] . nv[b,m,:])   — WMMA f32 16x16x4, one wave per tile
__global__ void k_graph(const float* __restrict__ nv, float* __restrict__ graph) {
    int bid = blockIdx.x;                 // (NP/16)^2 tiles
    int b   = blockIdx.y;
    int nt  = bid / (NP / 16), mt = bid % (NP / 16);
    int lane = threadIdx.x & 31;
    int half = lane >> 4;                 // 0 or 1
    int l16  = lane & 15;
    const float* base = nv + (long)b * NP * EP;
    int row = nt * 16 + l16;
    int col = mt * 16 + l16;
    v8f acc = {};
    for (int kb = 0; kb < EP; kb += 4) {
        int k0 = kb + half * 2;
        v2f a, bb;
        a.x  = base[(long)row * EP + k0];
        a.y  = base[(long)row * EP + k0 + 1];
        bb.x = base[(long)col * EP + k0];
        bb.y = base[(long)col * EP + k0 + 1];
        acc = __builtin_amdgcn_wmma_f32_16x16x4_f32(false, a, false, bb, (short)0, acc, false, false);
    }
    float* g = graph + (long)b * NP * NP;
    for (int r = 0; r < 8; ++r) {
        float v = acc[r];
        v = v > 0.0f ? v : 0.0f;
        g[(long)(nt * 16 + r + half * 8) * NP + mt * 16 + l16] = v;
    }
}

// dvec[b,n] = rowsum(graph)^-0.5 — one wave per row, shuffle reduction
__global__ void k_rowsum(const float* __restrict__ graph, float* __restrict__ dvec) {
    int row  = blockIdx.x;                // B*NP rows
    int lane = threadIdx.x & 31;
    int n = row % NP, b = row / NP;
    const float* g = graph + ((long)b * NP + n) * NP;
    float s = 0.0f;
    for (int m = lane; m < NP; m += 32) s += g[m];
    for (int off = 16; off > 0; off >>= 1) s += __shfl_down(s, off, 32);
    if (lane == 0) dvec[row] = (s > 1e-12f) ? rsqrtf(s) : 0.0f;
}

// xss[b,m,c] = xs[b,m,c] * d[m]   (column prescale for the L@xs GEMM)
__global__ void k_xss(const float* __restrict__ xs, const float* __restrict__ dvec,
                      float* __restrict__ xss) {
    long idx = (long)blockIdx.x * blockDim.x + threadIdx.x;  // B*NP*KP
    if (idx >= (long)B_ * NP * KP) return;
    long bm = idx / KP;
    int m = (int)(bm % NP);
    int b = (int)(bm / NP);
    xss[idx] = xs[idx] * dvec[b * NP + m];
}

// xg2[b,n,c] = d[n] * sum_m graph[n,m] * xss[b,m,c]
// block = 5 waves (one per 16-col tile of KP); graph strip + xss chunk staged in LDS,
// so the graph strip is read from HBM exactly once per block.
__global__ void k_xg2(const float* __restrict__ graph, const float* __restrict__ xss,
                      const float* __restrict__ dvec, float* __restrict__ xg2) {
    __shared__ float sG[16 * CHK];        // 3 KB
    __shared__ float sX[CHK * KP];        // 15 KB
    int nt  = blockIdx.x;                 // NP/16 strips
    int b   = blockIdx.y;
    int tid = threadIdx.x;                // 0..159 (5 waves)
    int ct   = tid >> 5;                  // wave id -> c tile
    int lane = tid & 31;
    int half = lane >> 4;
    int l16  = lane & 15;
    int col  = ct * 16 + l16;
    const float* g  = graph + (long)b * NP * NP;
    const float* xb = xss + (long)b * NP * KP;
    const float* dv = dvec + b * NP;
    v8f acc = {};
    for (int kc = 0; kc < NP; kc += CHK) {
        // cooperative stage of graph strip rows (contiguous 48-float runs)
        for (int e = tid; e < 16 * CHK; e += 160) {
            int r = e / CHK, k = e % CHK;
            sG[e] = g[(long)(nt * 16 + r) * NP + kc + k];
        }
        // cooperative stage of prescaled xs chunk (contiguous 80-float rows)
        for (int e = tid; e < CHK * KP; e += 160) {
            int r = e / KP, cc = e % KP;
            sX[e] = xb[(long)(kc + r) * KP + cc];
        }
        if (kc + CHK < NP) {  // prefetch next chunk (global_prefetch_b8)
            __builtin_prefetch(&g[(long)(nt * 16 + (tid & 15)) * NP + kc + CHK], 0, 1);
            __builtin_prefetch(&xb[(long)(kc + CHK + (tid % CHK)) * KP], 0, 1);
        }
        __syncthreads();
        for (int kb = 0; kb < CHK; kb += 4) {
            int k0 = kb + half * 2;
            v2f a, bb;
            a.x  = sG[l16 * CHK + k0];
            a.y  = sG[l16 * CHK + k0 + 1];
            bb.x = sX[k0 * KP + col];
            bb.y = sX[(k0 + 1) * KP + col];
            acc = __builtin_amdgcn_wmma_f32_16x16x4_f32(false, a, false, bb, (short)0, acc, false, false);
        }
        __syncthreads();
    }
    float* o = xg2 + (long)b * NP * KP;
    for (int r = 0; r < 8; ++r) {
        int m = nt * 16 + r + half * 8;
        o[(long)m * KP + col] = acc[r] * dv[m];
    }
}

// out[b,n,o] = act( sum_i xs*W0 + sum_i xg2*W1 + bias )
// one wave per (node, pair of o-tiles): A fragment reused for both o-tiles.
__global__ void k_pernode(const float* __restrict__ xs, const float* __restrict__ xg2,
                          const float* __restrict__ W, const float* __restrict__ Bv,
                          int O, int act, float* __restrict__ out) {
    int opairs = O / 32;
    int op = blockIdx.x % opairs;
    int n  = blockIdx.x / opairs;
    int lane = threadIdx.x & 31;
    int half = lane >> 4;
    int l16  = lane & 15;
    int brow = l16;                        // batch row (valid < B_)
    int oc0  = op * 32 + l16;
    int oc1  = oc0 + 16;
    const float* Wn = W + (long)n * 2 * KP * O;
    v8f acc0 = {}, acc1 = {};
    for (int sel = 0; sel < 2; ++sel) {
        const float* src = sel ? xg2 : xs;
        const float* Ws  = Wn + (long)sel * KP * O;
        for (int kb = 0; kb < KP; kb += 4) {
            int k0 = kb + half * 2;
            v2f a, b0, b1;
            a.x = (brow < B_) ? src[((long)brow * NP + n) * KP + k0]     : 0.0f;
            a.y = (brow < B_) ? src[((long)brow * NP + n) * KP + k0 + 1] : 0.0f;
            b0.x = Ws[(long)k0 * O + oc0];
            b0.y = Ws[(long)(k0 + 1) * O + oc0];
            b1.x = Ws[(long)k0 * O + oc1];
            b1.y = Ws[(long)(k0 + 1) * O + oc1];
            acc0 = __builtin_amdgcn_wmma_f32_16x16x4_f32(false, a, false, b0, (short)0, acc0, false, false);
            acc1 = __builtin_amdgcn_wmma_f32_16x16x4_f32(false, a, false, b1, (short)0, acc1, false, false);
        }
    }
    float bias0 = Bv[(long)n * O + oc0];
    float bias1 = Bv[(long)n * O + oc1];
    for (int r = 0; r < 8; ++r) {
        int m = r + half * 8;
        if (m < B_) {
            float v0 = acc0[r] + bias0;
            float v1 = acc1[r] + bias1;
            v0 = act ? tanhf(v0) : sigmoidf_(v0);
            v1 = act ? tanhf(v1) : sigmoidf_(v1);
            out[((long)m * N_ + n) * O + oc0] = v0;
            out[((long)m * N_ + n) * O + oc1] = v1;
        }
    }
}

// h = r*state + (1-r)*hc
__global__ void k_hupdate(const float* __restrict__ zr, const float* __restrict__ hc,
                          float* __restrict__ state) {
    long idx = (long)blockIdx.x * blockDim.x + threadIdx.x;  // B*N*H
    if (idx >= (long)B_ * N_ * H_) return;
    int h  = (int)(idx % H_);
    long bn = idx / H_;
    float r = zr[bn * OG + H_ + h];
    state[idx] = r * state[idx] + (1.0f - r) * hc[idx];
}

// predict[b,t,n] += ci[b,n,:] . predW[t,:] + predB[t]
__global__ void k_head(const float* __restrict__ state, const float* __restrict__ pW,
                       const float* __restrict__ pB, float* __restrict__ outp) {
    long idx = (long)blockIdx.x * blockDim.x + threadIdx.x;  // B*T*N
    if (idx >= (long)B_ * T_ * N_) return;
    int n  = (int)(idx % N_);
    int ot = (int)((idx / N_) % T_);
    int b  = (int)(idx / ((long)N_ * T_));
    const float* w = pW + ot * H_;
    const float* s = state + ((long)b * N_ + n) * H_;
    float a = pB[ot];
    for (int h = 0; h < H_; ++h) a += s[h] * w[h];
    outp[idx] += a;
}

// cur[t,b,n] -= ci[b,n,:] . skipW[t,:] + skipB[t]
__global__ void k_skip(const float* __restrict__ state, const float* __restrict__ sW,
                       const float* __restrict__ sB, float* __restrict__ cur) {
    long idx = (long)blockIdx.x * blockDim.x + threadIdx.x;  // T*B*N
    if (idx >= (long)T_ * B_ * N_) return;
    int n = (int)(idx % N_);
    int b = (int)((idx / N_) % B_);
    int t = (int)(idx / ((long)N_ * B_));
    const float* w = sW + t * H_;
    const float* s = state + ((long)b * N_ + n) * H_;
    float a = sB[t];
    for (int h = 0; h < H_; ++h) a += s[h] * w[h];
    cur[idx] -= a;
}

// ---------------------------------------------------------------- launcher
static inline dim3 g1(long n, int bs) { return dim3((unsigned)((n + bs - 1) / bs)); }

extern "C" void kernel_launch(void* const* d_in, const int* in_sizes, int n_in,
                              void* d_out, int out_size, void* d_ws, size_t ws_size,
                              hipStream_t stream) {
    const float* src    = (const float*)d_in[0];
    const float* nodeE  = (const float*)d_in[1];
    const float* tidE   = (const float*)d_in[2];
    const float* diwE   = (const float*)d_in[3];
    const float* wp_g   = (const float*)d_in[4];
    const float* bp_g   = (const float*)d_in[5];
    const float* fc1W_g = (const float*)d_in[6];
    const float* fc1b_g = (const float*)d_in[7];
    const float* fc2W_g = (const float*)d_in[8];
    const float* fc2b_g = (const float*)d_in[9];
    const float* fc3W_g = (const float*)d_in[10];
    const float* fc3b_g = (const float*)d_in[11];
    const float* wp_u   = (const float*)d_in[12];
    const float* bp_u   = (const float*)d_in[13];
    const float* fc1W_u = (const float*)d_in[14];
    const float* fc1b_u = (const float*)d_in[15];
    const float* fc2W_u = (const float*)d_in[16];
    const float* fc2b_u = (const float*)d_in[17];
    const float* fc3W_u = (const float*)d_in[18];
    const float* fc3b_u = (const float*)d_in[19];
    const float* predW  = (const float*)d_in[20];
    const float* predB  = (const float*)d_in[21];
    const float* skipW  = (const float*)d_in[22];
    const float* skipB  = (const float*)d_in[23];
    float* outp = (float*)d_out;
    (void)in_sizes; (void)n_in; (void)ws_size;

    // workspace carve-out (floats)
    float* p = (float*)d_ws;
    auto alloc = [&](size_t nfl) { float* r = p; p += nfl; return r; };
    float* emb   = alloc((size_t)B_ * T_ * N_ * E_);
    float* Wg    = alloc((size_t)N_ * 2 * KP * OG);
    float* Bg    = alloc((size_t)N_ * OG);
    float* Wu    = alloc((size_t)N_ * 2 * KP * OU);
    float* Bu    = alloc((size_t)N_ * OU);
    float* state = alloc((size_t)B_ * N_ * H_);
    float* xsb   = alloc((size_t)B_ * NP * KP);
    float* xss   = alloc((size_t)B_ * NP * KP);
    float* xg2   = alloc((size_t)B_ * NP * KP);
    float* nv    = alloc((size_t)B_ * NP * EP);
    float* graph = alloc((size_t)B_ * NP * NP);
    float* dvec  = alloc((size_t)B_ * NP);
    float* zr    = alloc((size_t)B_ * N_ * OG);
    float* hc    = alloc((size_t)B_ * N_ * H_);
    float* cur   = alloc((size_t)T_ * B_ * N_);

    k_embed<<<g1((long)B_ * T_ * N_, 256), 256, 0, stream>>>(src, nodeE, tidE, diwE, emb);
    k_init_cur<<<g1((long)T_ * B_ * N_, 256), 256, 0, stream>>>(src, cur);
    k_zero<<<g1(out_size, 256), 256, 0, stream>>>(outp, out_size);

    for (int c = 0; c < 2; ++c) {
        k_node_w<<<g1((long)N_ * 2 * KP * OG, 256), 256, 0, stream>>>(
            nodeE, wp_g + (size_t)c * E_ * 2 * DIN * OG, Wg, OG);
        k_node_b<<<g1((long)N_ * OG, 256), 256, 0, stream>>>(
            nodeE, bp_g + (size_t)c * E_ * OG, Bg, OG);
        k_node_w<<<g1((long)N_ * 2 * KP * OU, 256), 256, 0, stream>>>(
            nodeE, wp_u + (size_t)c * E_ * 2 * DIN * OU, Wu, OU);
        k_node_b<<<g1((long)N_ * OU, 256), 256, 0, stream>>>(
            nodeE, bp_u + (size_t)c * E_ * OU, Bu, OU);
        k_zero<<<g1((long)B_ * N_ * H_, 256), 256, 0, stream>>>(state, (long)B_ * N_ * H_);

        for (int t = 0; t < T_; ++t) {
            const float* curx = cur + (size_t)t * B_ * N_;
            // ---- gate GCN -> zr (sigmoid applied in k_pernode)
            k_build_xs<<<g1((long)B_ * NP * KP, 256), 256, 0, stream>>>(curx, state, zr, 0, xsb);
            k_fc<<<g1((long)B_ * NP, 128), 128, 0, stream>>>(
                xsb, emb, t,
                fc1W_g + (size_t)c * 16 * DIN, fc1b_g + c * 16,
                fc2W_g + (size_t)c * 2 * 16,   fc2b_g + c * 2,
                fc3W_g + (size_t)c * E_ * 2,   fc3b_g + c * E_, nv);
            k_graph<<<dim3((NP / 16) * (NP / 16), B_), 32, 0, stream>>>(nv, graph);
            k_rowsum<<<dim3(B_ * NP), 32, 0, stream>>>(graph, dvec);
            k_xss<<<g1((long)B_ * NP * KP, 256), 256, 0, stream>>>(xsb, dvec, xss);
            k_xg2<<<dim3(NP / 16, B_), 160, 0, stream>>>(graph, xss, dvec, xg2);
            k_pernode<<<dim3(N_ * (OG / 32)), 32, 0, stream>>>(xsb, xg2, Wg, Bg, OG, 0, zr);
            // ---- update GCN -> hc (tanh applied in k_pernode)
            k_build_xs<<<g1((long)B_ * NP * KP, 256), 256, 0, stream>>>(curx, state, zr, 1, xsb);
            k_fc<<<g1((long)B_ * NP, 128), 128, 0, stream>>>(
                xsb, emb, t,
                fc1W_u + (size_t)c * 16 * DIN, fc1b_u + c * 16,
                fc2W_u + (size_t)c * 2 * 16,   fc2b_u + c * 2,
                fc3W_u + (size_t)c * E_ * 2,   fc3b_u + c * E_, nv);
            k_graph<<<dim3((NP / 16) * (NP / 16), B_), 32, 0, stream>>>(nv, graph);
            k_rowsum<<<dim3(B_ * NP), 32, 0, stream>>>(graph, dvec);
            k_xss<<<g1((long)B_ * NP * KP, 256), 256, 0, stream>>>(xsb, dvec, xss);
            k_xg2<<<dim3(NP / 16, B_), 160, 0, stream>>>(graph, xss, dvec, xg2);
            k_pernode<<<dim3(N_ * (OU / 32)), 32, 0, stream>>>(xsb, xg2, Wu, Bu, OU, 1, hc);
            // ---- h = r*state + (1-r)*hc
            k_hupdate<<<g1((long)B_ * N_ * H_, 256), 256, 0, stream>>>(zr, hc, state);
        }

        k_head<<<g1((long)B_ * T_ * N_, 256), 256, 0, stream>>>(
            state, predW + (size_t)c * T_ * H_, predB + c * T_, outp);
        if (c == 0)
            k_skip<<<g1((long)T_ * B_ * N_, 256), 256, 0, stream>>>(state, skipW, skipB, cur);
    }
}